// SigmaifLayer_45268955300116
// MI455X (gfx1250) — compile-verified
//
#include <hip/hip_runtime.h>
#include <cstdint>

// ---- constants from the reference ----
#define IN_F   4096
#define OUT_F  4096
#define KGRP   16          // groups of 256 consecutive features
// THRESH == 0.0f

typedef __attribute__((ext_vector_type(2))) float v2f;
typedef __attribute__((ext_vector_type(4))) float f4;   // native vector -> nontemporal builtins OK
typedef __attribute__((ext_vector_type(8))) float v8f;

// ---------------------------------------------------------------------------
// Kernel 1: wsum[n] = sum_o W[o][n]   (== ones(1,4096) @ W)
// Implemented with V_WMMA_F32_16X16X4_F32, A = all-ones.
// C[m][n] += sum_k A[m][k]*B[k][n]; with A==1 this is a column sum of the
// 4x16 B tile, accumulated over all 1024 K-steps. Row M=0 of C (VGPR0,
// lanes 0..15) then holds the 16 column sums.
// ---------------------------------------------------------------------------
__global__ __launch_bounds__(256) void colsum_wmma_kernel(const float* __restrict__ W,
                                                          float* __restrict__ wsum) {
    const int lane  = threadIdx.x & 31;
    const int wave  = (blockIdx.x * blockDim.x + threadIdx.x) >> 5;
    const int n0    = wave * 16;                 // 16 columns per wave
    const int nn    = n0 + (lane & 15);          // B/C: N = lane & 15 (ISA layout)
    const int kbase = (lane >> 4) * 2;           // half-wave K split; exact slot
                                                 // irrelevant for a column sum
    const float* col = W + nn;

    v2f a; a.x = 1.0f; a.y = 1.0f;               // all-ones A matrix
    v8f c = {};

    #pragma unroll 4
    for (int k0 = 0; k0 < IN_F; k0 += 4) {
        v2f b;
        b.x = col[(size_t)(k0 + kbase + 0) * IN_F];
        b.y = col[(size_t)(k0 + kbase + 1) * IN_F];
        // (neg_a, A, neg_b, B, c_mod, C, reuse_a, reuse_b)
        c = __builtin_amdgcn_wmma_f32_16x16x4_f32(false, a, false, b,
                                                  (short)0, c, false, false);
    }
    if (lane < 16) wsum[nn] = c[0];              // M=0 row of C = column sums
}

// ---------------------------------------------------------------------------
// Kernel 2: per row -> 16 group sums of x*wsum, prefix scan, first >=0 pick,
// then out[row, j] = sigmoid(S + bias[j]).  One wave (32 lanes) per row.
// x read and out write are the 2x256MiB streams -> nontemporal 128b accesses.
// ---------------------------------------------------------------------------
__global__ __launch_bounds__(256) void sigmaif_main_kernel(const float* __restrict__ x,
                                                           const float* __restrict__ bias,
                                                           const float* __restrict__ wsum,
                                                           float* __restrict__ out) {
    __shared__ f4 sW[IN_F / 4];    // 16 KiB: wsum
    __shared__ f4 sB[OUT_F / 4];   // 16 KiB: bias

    const int t = threadIdx.x;
    const f4* wsum4 = (const f4*)wsum;
    const f4* bias4 = (const f4*)bias;
    #pragma unroll
    for (int i = 0; i < 4; ++i) {                // 256 threads x 4 = 1024 f4
        sW[t + i * 256] = wsum4[t + i * 256];
        sB[t + i * 256] = bias4[t + i * 256];
    }
    __syncthreads();

    const int lane   = t & 31;
    const int waveId = t >> 5;                   // 8 waves per block
    const int row    = blockIdx.x * 8 + waveId;

    const f4* xr = (const f4*)(x + (size_t)row * IN_F);

    // ---- group sums + scan ----
    float cs = 0.0f, S = 0.0f;
    int done = 0;
    #pragma unroll
    for (int g = 0; g < KGRP; ++g) {
        float acc = 0.0f;
        #pragma unroll
        for (int it = 0; it < 2; ++it) {         // 2 x (32 lanes x f4) = 256 floats
            const int i = g * 64 + it * 32 + lane;
            f4 v = __builtin_nontemporal_load(&xr[i]);
            f4 w = sW[i];
            acc += v.x * w.x + v.y * w.y + v.z * w.z + v.w * w.w;
        }
        // wave32 butterfly reduction -> every lane holds the group sum
        #pragma unroll
        for (int off = 16; off > 0; off >>= 1)
            acc += __shfl_xor(acc, off, 32);
        cs += acc;
        if (!done && (cs >= 0.0f || g == KGRP - 1)) { S = cs; done = 1; }
    }

    // ---- broadcast sigmoid row write ----
    f4* outr = (f4*)(out + (size_t)row * OUT_F);
    #pragma unroll 4
    for (int i = lane; i < OUT_F / 4; i += 32) {
        f4 b = sB[i];
        f4 o;
        o.x = 1.0f / (1.0f + __expf(-(S + b.x)));
        o.y = 1.0f / (1.0f + __expf(-(S + b.y)));
        o.z = 1.0f / (1.0f + __expf(-(S + b.z)));
        o.w = 1.0f / (1.0f + __expf(-(S + b.w)));
        __builtin_nontemporal_store(o, &outr[i]);
    }
}

// ---------------------------------------------------------------------------
extern "C" void kernel_launch(void* const* d_in, const int* in_sizes, int n_in,
                              void* d_out, int out_size, void* d_ws, size_t ws_size,
                              hipStream_t stream) {
    (void)n_in; (void)out_size; (void)ws_size;

    const float* x    = (const float*)d_in[0];   // [8,2048,4096] f32
    const float* W    = (const float*)d_in[1];   // [4096,4096]   f32
    const float* bias = (const float*)d_in[2];   // [4096]        f32
    float*       out  = (float*)d_out;           // [8,2048,4096] f32
    float*       wsum = (float*)d_ws;            // [4096] scratch

    // Kernel 1: 4096 columns / 16 per wave = 256 waves = 32 blocks x 8 waves
    colsum_wmma_kernel<<<32, 256, 0, stream>>>(W, wsum);

    // Kernel 2: one wave per row, 8 waves per block
    const int rows = in_sizes[0] / IN_F;         // 16384
    sigmaif_main_kernel<<<rows / 8, 256, 0, stream>>>(x, bias, wsum, out);
}